// SelfAttention_89172110999855
// MI455X (gfx1250) — compile-verified
//
#include <hip/hip_runtime.h>
#include <hip/hip_bf16.h>

typedef __attribute__((ext_vector_type(2))) float v2f;
typedef __attribute__((ext_vector_type(8))) float v8f;

#define BATCH 32
#define CDIM 1024
#define NHEAD 16
#define HDIM 64
#define TSEQ 4096

// ---------------------------------------------------------------------------
// GEMM: Y(32 x N) = X(32 x K) * W(K x N) + bias, f32, V_WMMA_F32_16X16X4_F32.
// One wave computes a 32x32 output patch = 2 M-tiles x 2 N-tiles, so the two
// M-tiles share every B load and the two N-tiles share every A load:
// per K-step: 2 x b64 (A) + 4 x b32 (B) feed 4 WMMAs.
// A layout (16x4 f32): lanes 0-15 hold M=lane, K=(0,1); lanes 16-31 K=(2,3)
// B layout (4x16 f32): lanes 0-15 hold K=(0,1) at N=lane; lanes 16-31 K=(2,3)
// C layout: VGPR r -> row r + 8*(lane>=16), col = lane&15
// ---------------------------------------------------------------------------
__global__ void gemm32_wmma(const float* __restrict__ X,
                            const float* __restrict__ W,
                            const float* __restrict__ bias,
                            float* __restrict__ Y,
                            int N, int K) {
    const int wid  = blockIdx.x * (blockDim.x >> 5) + (threadIdx.x >> 5);
    const int lane = threadIdx.x & 31;
    const int ng   = wid;                 // 32-column group
    if (ng * 32 >= N) return;

    const int l15  = lane & 15;
    const int half = lane >> 4;           // 0 or 1
    const int koff = half * 2;            // K sub-offset for this half-wave

    const float* __restrict__ Xr0 = X + (size_t)l15 * K + koff;        // M-tile 0
    const float* __restrict__ Xr1 = X + (size_t)(16 + l15) * K + koff; // M-tile 1
    const int n0 = ng * 32 + l15;         // N-tile 0 column
    const int n1 = n0 + 16;               // N-tile 1 column

    v8f c00 = {}, c01 = {}, c10 = {}, c11 = {};
#pragma unroll 4
    for (int k = 0; k < K; k += 4) {
        const v2f a0 = *(const v2f*)(Xr0 + k);   // contiguous pair -> b64
        const v2f a1 = *(const v2f*)(Xr1 + k);
        const float* __restrict__ Wr0 = W + (size_t)(k + koff) * N;
        const float* __restrict__ Wr1 = Wr0 + N;
        v2f b0, b1;
        b0.x = Wr0[n0]; b0.y = Wr1[n0];
        b1.x = Wr0[n1]; b1.y = Wr1[n1];
        c00 = __builtin_amdgcn_wmma_f32_16x16x4_f32(false, a0, false, b0, (short)0, c00, false, false);
        c01 = __builtin_amdgcn_wmma_f32_16x16x4_f32(false, a0, false, b1, (short)0, c01, false, false);
        c10 = __builtin_amdgcn_wmma_f32_16x16x4_f32(false, a1, false, b0, (short)0, c10, false, false);
        c11 = __builtin_amdgcn_wmma_f32_16x16x4_f32(false, a1, false, b1, (short)0, c11, false, false);
    }

    const float bn0 = bias[n0];
    const float bn1 = bias[n1];
#pragma unroll
    for (int r = 0; r < 8; ++r) {
        const int row0 = r + 8 * half;        // M-tile 0 rows 0..15
        const int row1 = 16 + row0;           // M-tile 1 rows 16..31
        Y[(size_t)row0 * N + n0] = c00[r] + bn0;
        Y[(size_t)row0 * N + n1] = c01[r] + bn1;
        Y[(size_t)row1 * N + n0] = c10[r] + bn0;
        Y[(size_t)row1 * N + n1] = c11[r] + bn1;
    }
}

// ---------------------------------------------------------------------------
// Copy k,v slices of qkv buffer (B x 3072) into d_out sections (B x 1024 each)
// ---------------------------------------------------------------------------
__global__ void copy_kv(const float* __restrict__ qkv, float* __restrict__ dst) {
    const int nthr = gridDim.x * blockDim.x;
    for (int e = blockIdx.x * blockDim.x + threadIdx.x; e < 2 * BATCH * CDIM; e += nthr) {
        const int sel = e >> 15;          // 0 = k, 1 = v
        const int r   = e & 32767;
        const int b   = r >> 10;
        const int j   = r & 1023;
        dst[e] = qkv[(size_t)b * 3072 + 1024 + (sel << 10) + j];
    }
}

// ---------------------------------------------------------------------------
// Decode attention: one block per (b,h), 256 threads (8 waves).
// Only rows t <= pos[b] are read (mask -> softmax zero-weights the rest),
// halving average KV traffic. Scores in LDS; two-pass stable softmax.
// global_prefetch_b8 keeps the HBM stream ahead of the loadcnt waits.
// ---------------------------------------------------------------------------
__global__ void attn_decode(const float* __restrict__ qkv,
                            const float* __restrict__ Kc,
                            const float* __restrict__ Vc,
                            const int* __restrict__ pos,
                            float* __restrict__ out) {
    const int b   = blockIdx.x >> 4;
    const int h   = blockIdx.x & 15;
    const int tid = threadIdx.x;

    __shared__ __align__(16) float sq[HDIM];
    __shared__ float ss[TSEQ];
    __shared__ float sred[8];
    __shared__ float sout[4][HDIM];

    if (tid < HDIM) sq[tid] = qkv[(size_t)b * 3072 + h * HDIM + tid];
    __syncthreads();

    const int L = pos[b] + 1;
    const float scale = 0.125f;  // 1/sqrt(64)
    const size_t rowStride = (size_t)NHEAD * HDIM;
    const float* __restrict__ Kb = Kc + (size_t)b * TSEQ * rowStride + (size_t)h * HDIM;
    const float* __restrict__ Vb = Vc + (size_t)b * TSEQ * rowStride + (size_t)h * HDIM;

    // ---- scores: one half-wave (16 lanes x float4 = one 256B K row) per t ----
    const int hw = tid >> 4;      // 0..15
    const int j  = tid & 15;
    const float4 q4 = ((const float4*)sq)[j];
    for (int t = hw; t < L; t += 16) {
        __builtin_prefetch(Kb + (size_t)(t + 32) * rowStride + j * 4, 0, 0);
        const float4 k4 = *(const float4*)(Kb + (size_t)t * rowStride + j * 4);
        float p = k4.x * q4.x + k4.y * q4.y + k4.z * q4.z + k4.w * q4.w;
        p += __shfl_xor(p, 1, 32);
        p += __shfl_xor(p, 2, 32);
        p += __shfl_xor(p, 4, 32);
        p += __shfl_xor(p, 8, 32);
        if (j == 0) ss[t] = p * scale;
    }
    __syncthreads();

    // ---- max reduction ----
    float m = -3.402823466e38f;
    for (int t = tid; t < L; t += 256) m = fmaxf(m, ss[t]);
#pragma unroll
    for (int o = 16; o > 0; o >>= 1) m = fmaxf(m, __shfl_xor(m, o, 32));
    if ((tid & 31) == 0) sred[tid >> 5] = m;
    __syncthreads();
    m = sred[0];
#pragma unroll
    for (int w = 1; w < 8; ++w) m = fmaxf(m, sred[w]);

    // ---- exp + sum ----
    float s = 0.f;
    for (int t = tid; t < L; t += 256) {
        const float e = __expf(ss[t] - m);
        ss[t] = e;
        s += e;
    }
#pragma unroll
    for (int o = 16; o > 0; o >>= 1) s += __shfl_xor(s, o, 32);
    __syncthreads();                 // everyone done reading sred (max)
    if ((tid & 31) == 0) sred[tid >> 5] = s;
    __syncthreads();
    s = 0.f;
#pragma unroll
    for (int w = 0; w < 8; ++w) s += sred[w];
    const float inv = 1.0f / s;

    // ---- out = sum_t p[t] * V[t] : 64 lanes cover one 256B V row, 4 t-groups ----
    const int d = tid & 63;
    const int g = tid >> 6;          // 0..3
    float acc = 0.f;
    for (int t = g; t < L; t += 4) {
        __builtin_prefetch(Vb + (size_t)(t + 8) * rowStride + d, 0, 0);
        acc += ss[t] * Vb[(size_t)t * rowStride + d];
    }
    sout[g][d] = acc;
    __syncthreads();
    if (tid < HDIM) {
        const float r = (sout[0][tid] + sout[1][tid] + sout[2][tid] + sout[3][tid]) * inv;
        out[(size_t)b * CDIM + h * HDIM + tid] = r;
    }
}

extern "C" void kernel_launch(void* const* d_in, const int* in_sizes, int n_in,
                              void* d_out, int out_size, void* d_ws, size_t ws_size,
                              hipStream_t stream) {
    (void)in_sizes; (void)n_in; (void)out_size; (void)ws_size;

    const float* x      = (const float*)d_in[0];
    const float* prev_k = (const float*)d_in[1];
    const float* prev_v = (const float*)d_in[2];
    const int*   pos    = (const int*)d_in[3];
    const float* w_attn = (const float*)d_in[4];
    const float* b_attn = (const float*)d_in[5];
    const float* w_proj = (const float*)d_in[6];
    const float* b_proj = (const float*)d_in[7];
    float* out = (float*)d_out;

    float* qkvbuf  = (float*)d_ws;                  // 32*3072 f32
    float* attnout = qkvbuf + BATCH * 3 * CDIM;     // 32*1024 f32

    // 1) QKV GEMM: 3072/32 = 96 waves -> 24 blocks x 4 waves
    gemm32_wmma<<<24, 128, 0, stream>>>(x, w_attn, b_attn, qkvbuf, 3 * CDIM, CDIM);

    // 2) route k,v into d_out (after the out section of 32768 floats)
    copy_kv<<<64, 256, 0, stream>>>(qkvbuf, out + BATCH * CDIM);

    // 3) attention decode: one block per (b,h)
    attn_decode<<<BATCH * NHEAD, 256, 0, stream>>>(qkvbuf, prev_k, prev_v, pos, attnout);

    // 4) output projection: 1024/32 = 32 waves -> 8 blocks x 4 waves
    gemm32_wmma<<<8, 128, 0, stream>>>(attnout, w_proj, b_proj, out, CDIM, CDIM);
}